// RGCN_1202590843052
// MI455X (gfx1250) — compile-verified
//
#include <hip/hip_runtime.h>
#include <hip/hip_bf16.h>

#define N_NODES 50000
#define N_EDGES 800000
#define R_REL   16
#define C_CH    128
#define K_REL   (R_REL * C_CH)          // 2048
#define K_EXT   (K_REL + C_CH)          // 2176 (relations + root)
#define BN_EPS  1e-5f

typedef float v2f __attribute__((ext_vector_type(2)));
typedef float v8f __attribute__((ext_vector_type(8)));

// ---------------------------------------------------------------------------
// Scatter: per-edge wave32. Gather x[src] (128 f32 = float4/lane), atomic-add
// into sums[dst*R + rel], count edges per segment.
// ---------------------------------------------------------------------------
__global__ __launch_bounds__(256)
void rgcn_scatter_kernel(const int* __restrict__ src, const int* __restrict__ dst,
                         const int* __restrict__ etype, const float* __restrict__ x,
                         float* __restrict__ sums, float* __restrict__ cnt) {
    int e    = blockIdx.x * (blockDim.x >> 5) + (threadIdx.x >> 5);
    int lane = threadIdx.x & 31;
    if (e >= N_EDGES) return;
    int s = src[e];
    int d = dst[e];
    int r = etype[e];
    r = (r > R_REL - 1) ? (R_REL - 1) : r;         // clamp(max=R-1)
    int seg = d * R_REL + r;

    float4 v = ((const float4*)x)[(long long)s * (C_CH / 4) + lane];
    float* base = sums + (long long)seg * C_CH + lane * 4;
    atomicAdd(base + 0, v.x);
    atomicAdd(base + 1, v.y);
    atomicAdd(base + 2, v.z);
    atomicAdd(base + 3, v.w);
    if (lane == 0) atomicAdd(cnt + seg, 1.0f);
}

// cnt[i] <- 1 / max(cnt[i], 1)   (fold mean into A-tile load)
__global__ void rgcn_invcnt_kernel(float* __restrict__ cnt) {
    int i = blockIdx.x * blockDim.x + threadIdx.x;
    if (i < N_NODES * R_REL) cnt[i] = 1.0f / fmaxf(cnt[i], 1.0f);
}

// ---------------------------------------------------------------------------
// Fused GEMM: out[n,o] = sum_k A_ext[n,k] * B_ext[k,o] + bias[o]
//   A_ext[n, r*128+i] = sums[n,r,i] * inv[n,r]   (k < 2048)
//   A_ext[n, 2048+i]  = xr[n,i]                  (root term)
//   B_ext[r*128+i, o] = W[r,i,o];  B_ext[2048+i, o] = root[i,o]
// Block: 256 threads (8 waves) -> 128x128 output tile; K chunked by 32 in LDS.
// Matrix op: V_WMMA_F32_16X16X4_F32 (exact fp32, 16x16 tile per wave strip).
// ---------------------------------------------------------------------------
__global__ __launch_bounds__(256)
void rgcn_gemm_kernel(const float* __restrict__ sums, const float* __restrict__ inv,
                      const float* __restrict__ xr, const float* __restrict__ W,
                      const float* __restrict__ root, const float* __restrict__ bias,
                      float* __restrict__ out) {
    __shared__ float As[128 * 33];   // 128 rows x 32 k, stride 33 (odd -> conflict-free)
    __shared__ float Bs[32 * 132];   // 32 k x 128 cols, stride 132

    const int tid  = threadIdx.x;
    const int lane = tid & 31;
    const int wave = tid >> 5;
    const int row0 = blockIdx.x * 128;

    v8f acc[8];
    #pragma unroll
    for (int i = 0; i < 8; ++i) acc[i] = (v8f){0.f, 0.f, 0.f, 0.f, 0.f, 0.f, 0.f, 0.f};

    const int cb   = wave * 16;             // this wave's 16-column strip
    const int m    = lane & 15;             // row/col within 16x16 tile
    const int koff = (lane >> 4) << 1;      // half-wave K offset: 0 or 2 (ISA 16x4 f32 layout)

    // A-tile loader coords: 8 threads x float4 cover one 32-float row
    const int atk = (tid & 7) << 2;
    const int atr = tid >> 3;
    // B-tile loader coords: 32 threads x float4 cover one 128-float row
    const int btc = (tid & 31) << 2;
    const int btk = tid >> 5;

    for (int k0 = 0; k0 < K_EXT; k0 += 32) {
        // ---- stage A chunk (apply mean scaling on the fly) ----
        #pragma unroll
        for (int rr = 0; rr < 128; rr += 32) {
            int n = row0 + atr + rr;
            if (n > N_NODES - 1) n = N_NODES - 1;       // clamp, store is guarded
            int kg = k0 + atk;
            float4 v;
            if (kg < K_REL) {
                const float* ap = sums + (long long)n * K_REL + kg;
                v = *(const float4*)ap;
                float sc = inv[n * R_REL + (kg >> 7)];  // relation = kg/128
                v.x *= sc; v.y *= sc; v.z *= sc; v.w *= sc;
                if (k0 + 32 < K_REL)                    // hint next chunk
                    __builtin_prefetch(ap + 32, 0, 1);
            } else {
                v = *(const float4*)(xr + (long long)n * C_CH + (kg - K_REL));
            }
            float* p = &As[(atr + rr) * 33 + atk];
            p[0] = v.x; p[1] = v.y; p[2] = v.z; p[3] = v.w;
        }
        // ---- stage B chunk ----
        #pragma unroll
        for (int kk = 0; kk < 32; kk += 8) {
            int kg = k0 + btk + kk;
            const float* sp = (kg < K_REL) ? (W + (long long)kg * C_CH + btc)
                                           : (root + (long long)(kg - K_REL) * C_CH + btc);
            float4 v = *(const float4*)sp;
            float* p = &Bs[(btk + kk) * 132 + btc];
            p[0] = v.x; p[1] = v.y; p[2] = v.z; p[3] = v.w;
        }
        __syncthreads();

        // ---- 8 k-steps of 4, 8 M-tiles each: v_wmma_f32_16x16x4_f32 ----
        #pragma unroll
        for (int kk = 0; kk < 32; kk += 4) {
            v2f b;
            b.x = Bs[(kk + koff    ) * 132 + cb + m];
            b.y = Bs[(kk + koff + 1) * 132 + cb + m];
            #pragma unroll
            for (int mt = 0; mt < 8; ++mt) {
                v2f a;
                a.x = As[(mt * 16 + m) * 33 + kk + koff];
                a.y = As[(mt * 16 + m) * 33 + kk + koff + 1];
                acc[mt] = __builtin_amdgcn_wmma_f32_16x16x4_f32(
                    false, a, false, b, (short)0, acc[mt], false, false);
            }
        }
        __syncthreads();
    }

    // ---- store: D VGPR j -> row (j + 8*halfwave), col = cb + (lane&15) ----
    const int col   = cb + m;
    const float bv  = bias[col];
    const int mbase = (lane >> 4) << 3;
    #pragma unroll
    for (int mt = 0; mt < 8; ++mt) {
        #pragma unroll
        for (int j = 0; j < 8; ++j) {
            int row = row0 + mt * 16 + mbase + j;
            if (row < N_NODES)
                out[(long long)row * C_CH + col] = acc[mt][j] + bv;
        }
    }
}

// ---------------------------------------------------------------------------
// BatchNorm: per-channel sum / sumsq reduction, then fused norm+affine+ReLU.
// ---------------------------------------------------------------------------
__global__ __launch_bounds__(256)
void bn_stats_kernel(const float* __restrict__ h, float* __restrict__ stats) {
    __shared__ float sh_s[256];
    __shared__ float sh_q[256];
    int c    = threadIdx.x & 127;
    int half = threadIdx.x >> 7;
    float s = 0.f, q = 0.f;
    for (int n = blockIdx.x * 2 + half; n < N_NODES; n += gridDim.x * 2) {
        float v = h[(long long)n * C_CH + c];
        s += v; q += v * v;
    }
    sh_s[threadIdx.x] = s;
    sh_q[threadIdx.x] = q;
    __syncthreads();
    if (half == 0) {
        atomicAdd(&stats[c],        sh_s[c] + sh_s[c + 128]);
        atomicAdd(&stats[128 + c],  sh_q[c] + sh_q[c + 128]);
    }
}

__global__ __launch_bounds__(256)
void bn_relu_kernel(const float* __restrict__ stats, const float* __restrict__ bnw,
                    const float* __restrict__ bnb, float* __restrict__ h) {
    long long i = (long long)blockIdx.x * blockDim.x + threadIdx.x;
    if (i >= (long long)N_NODES * C_CH) return;
    int c = (int)(i & (C_CH - 1));
    const float invN = 1.0f / (float)N_NODES;
    float mu  = stats[c] * invN;
    float var = stats[128 + c] * invN - mu * mu;   // biased variance
    float g   = bnw[c] * rsqrtf(var + BN_EPS);
    float v   = (h[i] - mu) * g + bnb[c];
    h[i] = v > 0.f ? v : 0.f;
}

// ---------------------------------------------------------------------------
extern "C" void kernel_launch(void* const* d_in, const int* in_sizes, int n_in,
                              void* d_out, int out_size, void* d_ws, size_t ws_size,
                              hipStream_t stream) {
    (void)in_sizes; (void)n_in; (void)out_size; (void)ws_size;
    const float* x     = (const float*)d_in[0];
    const int*   eidx  = (const int*)d_in[1];    // [2,E]
    const float* eattr = (const float*)d_in[2];  // [E,8]
    const int*   etype = (const int*)d_in[3];
    const float* W1    = (const float*)d_in[4];
    const float* root1 = (const float*)d_in[5];
    const float* b1    = (const float*)d_in[6];
    const float* bnw   = (const float*)d_in[7];
    const float* bnb   = (const float*)d_in[8];
    const float* W2    = (const float*)d_in[9];
    const float* root2 = (const float*)d_in[10];
    const float* b2    = (const float*)d_in[11];

    // workspace layout (floats): cnt[N*R] | sums[N*R*C] | h[N*C] | stats[256]
    float* cnt   = (float*)d_ws;
    float* sums  = cnt  + (size_t)N_NODES * R_REL;
    float* h     = sums + (size_t)N_NODES * R_REL * C_CH;
    float* stats = h    + (size_t)N_NODES * C_CH;
    float* out   = (float*)d_out;

    const int* src = eidx;
    const int* dst = eidx + N_EDGES;

    const size_t segBytes =
        ((size_t)N_NODES * R_REL + (size_t)N_NODES * R_REL * C_CH) * sizeof(float);

    dim3 sBlk(256), sGrd((N_EDGES + 7) / 8);            // 8 edges (waves) per block
    dim3 gBlk(256), gGrd((N_NODES + 127) / 128);        // 128-row GEMM tiles
    dim3 iGrd((N_NODES * R_REL + 255) / 256);
    dim3 eGrd((unsigned)(((size_t)N_NODES * C_CH + 255) / 256));

    // ---- layer 1 ----
    hipMemsetAsync(d_ws, 0, segBytes, stream);
    rgcn_scatter_kernel<<<sGrd, sBlk, 0, stream>>>(src, dst, etype, x, sums, cnt);
    rgcn_invcnt_kernel<<<iGrd, 256, 0, stream>>>(cnt);
    rgcn_gemm_kernel<<<gGrd, gBlk, 0, stream>>>(sums, cnt, x, W1, root1, b1, h);

    // ---- batchnorm + relu ----
    hipMemsetAsync(stats, 0, 256 * sizeof(float), stream);
    bn_stats_kernel<<<256, 256, 0, stream>>>(h, stats);
    bn_relu_kernel<<<eGrd, 256, 0, stream>>>(stats, bnw, bnb, h);

    // ---- layer 2 ----
    hipMemsetAsync(d_ws, 0, segBytes, stream);
    rgcn_scatter_kernel<<<sGrd, sBlk, 0, stream>>>(src, dst, etype, h, sums, cnt);
    rgcn_invcnt_kernel<<<iGrd, 256, 0, stream>>>(cnt);
    rgcn_gemm_kernel<<<gGrd, gBlk, 0, stream>>>(sums, cnt, h, W2, root2, b2, out);

    // ---- tuple output: append edge_attr passthrough ----
    hipMemcpyAsync(out + (size_t)N_NODES * C_CH, eattr,
                   (size_t)N_EDGES * 8 * sizeof(float),
                   hipMemcpyDeviceToDevice, stream);
}